// e2edro_40596030881885
// MI455X (gfx1250) — compile-verified
//
#include <hip/hip_runtime.h>
#include <math.h>

// Problem constants (from reference setup_inputs)
#define KDIM   512
#define NYD    1024
#define NOBS   4096
#define MROWS  4097      // NOBS + 1
#define NITERS 2000
#define LR0    0.5f

// Solver launch geometry
#define NB   128         // persistent blocks (device-wide barrier participants)
#define TPB  256         // 8 wave32 per block

typedef float v2f __attribute__((ext_vector_type(2)));
typedef float v8f __attribute__((ext_vector_type(8)));

#if __has_builtin(__builtin_amdgcn_wmma_f32_16x16x4_f32)
#define HAVE_WMMA_F32 1
#endif

// ---------------------------------------------------------------------------
// Kernel 1: Y_hat = X @ W^T + b  (fp32 WMMA 16x16x4), fused epilogue:
//   rows < 4096 : ep[row][col]  = Y[row][col] - Y_hat[row][col]
//   row == 4096 : y_hat[col]    = Y_hat[row][col]   (to d_out+1024 and ws)
// Each wave computes a 16x64 strip (A-fragment reused across 4 N-tiles).
// ---------------------------------------------------------------------------
__global__ __launch_bounds__(256) void gemm_resid_kernel(
    const float* __restrict__ X, const float* __restrict__ Y,
    const float* __restrict__ W, const float* __restrict__ bias,
    float* __restrict__ ep, float* __restrict__ yhat_ws,
    float* __restrict__ out_yhat)
{
  const int wid  = blockIdx.x * 8 + (threadIdx.x >> 5);   // 4112 wave-tasks
  const int lane = threadIdx.x & 31;
  const int half = lane >> 4;        // 0: lanes 0-15, 1: lanes 16-31
  const int l16  = lane & 15;
  const int mt = wid % 257;          // 257 M tiles (16 rows, last partial)
  const int ns = wid / 257;          // 16 N strips of 64 cols
  const int m0 = mt * 16;
  const int n0 = ns * 64;

  v8f acc[4] = {};   // 4 adjacent 16x16 C tiles, f32 accum

#if defined(HAVE_WMMA_F32)
  // A 16x4 f32 layout: lanes 0-15 -> M=0..15; VGPR0 = K=2*half, VGPR1 = K=2*half+1
  const int arow = m0 + l16;
  const bool avalid = (arow < MROWS);
  const float* xrow = X + (size_t)arow * KDIM;
  for (int k = 0; k < KDIM; k += 4) {
    v2f a; a.x = 0.f; a.y = 0.f;
    if (avalid) {
      const float2 t = *(const float2*)(xrow + k + 2 * half);
      a.x = t.x; a.y = t.y;
    }
    // B = W^T tile: B[kk][n] = W[n0+n][k+kk]; 4x16 layout mirrors A.
#pragma unroll
    for (int q = 0; q < 4; ++q) {
      const int col = n0 + q * 16 + l16;
      const float2 t = *(const float2*)(W + (size_t)col * KDIM + k + 2 * half);
      v2f bf; bf.x = t.x; bf.y = t.y;
      acc[q] = __builtin_amdgcn_wmma_f32_16x16x4_f32(
          /*neg_a=*/false, a, /*neg_b=*/false, bf,
          /*c_mod=*/(short)0, acc[q], /*reuse_a=*/false, /*reuse_b=*/false);
    }
  }
#else
  // VALU fallback matching the C/D lane layout (only if builtin missing)
#pragma unroll
  for (int q = 0; q < 4; ++q) {
    const int col = n0 + q * 16 + l16;
#pragma unroll
    for (int e = 0; e < 8; ++e) {
      const int row = m0 + e + 8 * half;
      float s = 0.f;
      if (row < MROWS)
        for (int k = 0; k < KDIM; ++k)
          s = fmaf(X[(size_t)row * KDIM + k], W[(size_t)col * KDIM + k], s);
      acc[q][e] = s;
    }
  }
#endif

  // Epilogue per C/D layout: VGPR e -> M = e + 8*half, N = lane%16
#pragma unroll
  for (int q = 0; q < 4; ++q) {
    const int col = n0 + q * 16 + l16;
    const float bv = bias[col];
#pragma unroll
    for (int e = 0; e < 8; ++e) {
      const int row = m0 + e + 8 * half;
      const float val = acc[q][e] + bv;
      if (row < NOBS) {
        ep[(size_t)row * NYD + col] = Y[(size_t)row * NYD + col] - val;
      } else if (row == NOBS) {
        out_yhat[col] = val;
        yhat_ws[col]  = val;
      }
    }
  }
}

// ---------------------------------------------------------------------------
// Kernel 2: 32x32 LDS-tiled transpose: epT (1024 x 4096) = ep^T
// ---------------------------------------------------------------------------
__global__ __launch_bounds__(256) void transpose_kernel(
    const float* __restrict__ ep, float* __restrict__ epT)
{
  __shared__ float tile[32][33];
  const int tx = threadIdx.x & 31;
  const int ty = threadIdx.x >> 5;               // 0..7
  const int x  = blockIdx.x * 32 + tx;           // ep col
  const int y0 = blockIdx.y * 32;                // ep row base
#pragma unroll
  for (int j = 0; j < 32; j += 8)
    tile[ty + j][tx] = ep[(size_t)(y0 + ty + j) * NYD + x];
  __syncthreads();
  const int x2 = y0 + tx;                        // epT col (= ep row)
  const int y2 = blockIdx.x * 32;                // epT row (= ep col)
#pragma unroll
  for (int j = 0; j < 32; j += 8)
    epT[(size_t)(y2 + ty + j) * NOBS + x2] = tile[tx][ty + j];
}

// ---------------------------------------------------------------------------
// Persistent solver helpers
// ---------------------------------------------------------------------------
__device__ __forceinline__ float wave_sum(float s) {
#pragma unroll
  for (int off = 16; off > 0; off >>= 1) s += __shfl_xor(s, off, 32);
  return s;
}

__device__ __forceinline__ void grid_barrier(unsigned* cnt, unsigned* gen) {
  __syncthreads();
  if (threadIdx.x == 0) {
    __threadfence();
    const unsigned g =
        __hip_atomic_load(gen, __ATOMIC_RELAXED, __HIP_MEMORY_SCOPE_AGENT);
    const unsigned arrived =
        __hip_atomic_fetch_add(cnt, 1u, __ATOMIC_ACQ_REL, __HIP_MEMORY_SCOPE_AGENT);
    if (arrived == (unsigned)(NB - 1)) {
      __hip_atomic_store(cnt, 0u, __ATOMIC_RELAXED, __HIP_MEMORY_SCOPE_AGENT);
      __hip_atomic_fetch_add(gen, 1u, __ATOMIC_RELEASE, __HIP_MEMORY_SCOPE_AGENT);
    } else {
      while (__hip_atomic_load(gen, __ATOMIC_ACQUIRE,
                               __HIP_MEMORY_SCOPE_AGENT) == g)
        __builtin_amdgcn_s_sleep(2);
    }
    __threadfence();
  }
  __syncthreads();
}

// ---------------------------------------------------------------------------
// Kernel 3: persistent TV-DRO projected-subgradient solver.
// 2 grid barriers / iteration; z, c, eta carried redundantly per block.
// ---------------------------------------------------------------------------
__global__ __launch_bounds__(TPB) void solve_kernel(
    const float* __restrict__ ep, const float* __restrict__ epT,
    float* __restrict__ r_g, float* __restrict__ v_g,
    const float* __restrict__ yhat, float* __restrict__ part,
    unsigned* __restrict__ bar, const float* __restrict__ delta_p,
    const float* __restrict__ gamma_p, float* __restrict__ z_out)
{
  __shared__ float z_sh[NYD];     // current z (also unsorted v in phase D)
  __shared__ float w_sh[NOBS];    // per-scenario gz weights
  __shared__ float u_sh[NYD];     // sort buffer
  __shared__ float css_sh[NYD];   // cumulative sums
  __shared__ float rloc_sh[32];   // this block's r values
  __shared__ float wt_sh[8];
  __shared__ float sc_sh[8];
  __shared__ int   rho_sh;

  const int   blk  = blockIdx.x;
  const int   tx   = threadIdx.x;
  const int   wv   = tx >> 5;
  const int   lane = tx & 31;
  const float delta = delta_p[0];
  const float gamma = gamma_p[0];
  const float invN  = 1.0f / (float)NOBS;
  unsigned* cnt = bar;
  unsigned* gen = bar + 1;

  // ---- init: z = 1/NY
  for (int j = tx; j < NYD; j += TPB) z_sh[j] = 1.0f / (float)NYD;
  __syncthreads();

  float c = 0.f, eta = 0.f;

  // ---- r0 = ep @ z0 ; c0 = mean(r0); eta0 = mean(r0^2) - c0^2
  {
#pragma unroll
    for (int rr = 0; rr < 4; ++rr) {
      const int row = blk * 32 + wv * 4 + rr;
      const float4* er = (const float4*)(ep + (size_t)row * NYD);
      const float4* z4 = (const float4*)z_sh;
      float s = 0.f;
      for (int jj = lane; jj < NYD / 4; jj += 32) {
        const float4 e = er[jj]; const float4 zz = z4[jj];
        s += e.x * zz.x + e.y * zz.y + e.z * zz.z + e.w * zz.w;
      }
      s = wave_sum(s);
      if (lane == 0) { r_g[row] = s; rloc_sh[wv * 4 + rr] = s; }
    }
    __syncthreads();
    if (tx < 32) {
      const float rv = rloc_sh[tx];
      float s1 = wave_sum(rv);
      float s2 = wave_sum(rv * rv);
      if (tx == 0) { float* p = part + blk * 8; p[0] = s1; p[1] = s2; }
    }
    grid_barrier(cnt, gen);
    if (tx < 32) {
      float s1 = 0.f, s2 = 0.f;
      for (int p = tx; p < NB; p += 32) { s1 += part[p * 8]; s2 += part[p * 8 + 1]; }
      s1 = wave_sum(s1); s2 = wave_sum(s2);
      if (tx == 0) { sc_sh[0] = s1; sc_sh[1] = s2; }
    }
    __syncthreads();
    c   = sc_sh[0] * invN;
    eta = sc_sh[1] * invN - c * c;
    grid_barrier(cnt, gen);   // protect partials before loop reuses them
  }

  // ---- main loop
  for (int t = 0; t < NITERS; ++t) {
    // Phase A: r = ep @ z (rows of this block) + per-block partial stats
#pragma unroll
    for (int rr = 0; rr < 4; ++rr) {
      const int row = blk * 32 + wv * 4 + rr;
      const float4* er = (const float4*)(ep + (size_t)row * NYD);
      const float4* z4 = (const float4*)z_sh;
      float s = 0.f;
      for (int jj = lane; jj < NYD / 4; jj += 32) {
        const float4 e = er[jj]; const float4 zz = z4[jj];
        s += e.x * zz.x + e.y * zz.y + e.z * zz.z + e.w * zz.w;
      }
      s = wave_sum(s);
      if (lane == 0) { r_g[row] = s; rloc_sh[wv * 4 + rr] = s; }
    }
    __syncthreads();
    if (tx < 32) {
      const float rv   = rloc_sh[tx];
      const float rc   = rv - c;
      const float risk = rc * rc;
      const float d    = risk - eta;
      float mmax = risk, mrc = rc;
      int   midx = blk * 32 + tx;
      float srelu = fmaxf(d, 0.f);
      float sind  = (d > 0.f) ? 1.f : 0.f;
      float srcv  = sind * rc;
#pragma unroll
      for (int off = 16; off > 0; off >>= 1) {
        const float om  = __shfl_xor(mmax, off, 32);
        const float orc = __shfl_xor(mrc, off, 32);
        const int   oi  = __shfl_xor(midx, off, 32);
        if (om > mmax || (om == mmax && oi < midx)) { mmax = om; mrc = orc; midx = oi; }
        srelu += __shfl_xor(srelu, off, 32);
        sind  += __shfl_xor(sind,  off, 32);
        srcv  += __shfl_xor(srcv,  off, 32);
      }
      if (tx == 0) {
        float* p = part + blk * 8;
        p[0] = mmax; p[1] = mrc; p[2] = __int_as_float(midx);
        p[3] = srelu; p[4] = sind; p[5] = srcv;
      }
    }
    grid_barrier(cnt, gen);

    // Phase B (redundant per block): combine partials, compute scalars
    if (tx < 32) {
      float mmax = -3.4e38f, mrc = 0.f; int midx = 0x7fffffff;
      float s1 = 0.f, s2 = 0.f, s3 = 0.f;
      for (int p = tx; p < NB; p += 32) {
        const float* pp = part + p * 8;
        const float om = pp[0], orc = pp[1];
        const int oi = __float_as_int(pp[2]);
        if (om > mmax || (om == mmax && oi < midx)) { mmax = om; mrc = orc; midx = oi; }
        s1 += pp[3]; s2 += pp[4]; s3 += pp[5];
      }
#pragma unroll
      for (int off = 16; off > 0; off >>= 1) {
        const float om  = __shfl_xor(mmax, off, 32);
        const float orc = __shfl_xor(mrc, off, 32);
        const int   oi  = __shfl_xor(midx, off, 32);
        if (om > mmax || (om == mmax && oi < midx)) { mmax = om; mrc = orc; midx = oi; }
        s1 += __shfl_xor(s1, off, 32);
        s2 += __shfl_xor(s2, off, 32);
        s3 += __shfl_xor(s3, off, 32);
      }
      if (tx == 0) {
        sc_sh[0] = mmax; sc_sh[1] = mrc; sc_sh[2] = __int_as_float(midx);
        sc_sh[3] = s1; sc_sh[4] = s2; sc_sh[5] = s3;
      }
    }
    __syncthreads();
    const float rmax  = sc_sh[0];
    const float rcmax = sc_sh[1];
    const int   imax  = __float_as_int(sc_sh[2]);
    const float Sind  = sc_sh[4];
    const float Src   = sc_sh[5];
    const float lamInd  = (rmax - eta > 0.f) ? 1.f : 0.f;
    const float gc      = -2.f * delta * lamInd * rcmax - 2.f * invN * Src;
    const float ge      = 1.f - delta * lamInd - Sind * invN;
    const float coefMax = 2.f * delta * lamInd * rcmax;
    const float lr      = LR0 / sqrtf((float)(t + 1));
    const float c_old = c, eta_old = eta;

    // Phase C: w in LDS, then gz per column (one wave per column); v = z - lr*gz
    for (int i = tx; i < NOBS; i += TPB) {
      const float rv = r_g[i];
      const float rc = rv - c_old;
      const float ind = ((rc * rc - eta_old) > 0.f) ? 1.f : 0.f;
      float wi = 2.f * invN * ind * rc;
      if (i == imax) wi += coefMax;
      w_sh[i] = wi;
    }
    __syncthreads();
    {
      const int col = blk * 8 + wv;
      const float4* et = (const float4*)(epT + (size_t)col * NOBS);
      const float4* w4 = (const float4*)w_sh;
      float s = 0.f;
      for (int jj = lane; jj < NOBS / 4; jj += 32) {
        const float4 e = et[jj]; const float4 ww = w4[jj];
        s += e.x * ww.x + e.y * ww.y + e.z * ww.z + e.w * ww.w;
      }
      s = wave_sum(s);
      if (lane == 0) {
        const float gz = s - gamma * yhat[col];
        v_g[col] = z_sh[col] - lr * gz;
      }
    }
    c   = c_old  - lr * gc;
    eta = eta_old - lr * ge;
    grid_barrier(cnt, gen);

    // Phase D (redundant per block): simplex projection of v -> z_sh
    for (int j = tx; j < NYD; j += TPB) {
      const float vj = v_g[j];
      z_sh[j] = vj;   // unsorted copy
      u_sh[j] = vj;   // to be sorted descending
    }
    if (tx == 0) rho_sh = 0;
    __syncthreads();
    // bitonic sort, descending
    for (unsigned k = 2; k <= NYD; k <<= 1) {
      for (unsigned j = k >> 1; j > 0; j >>= 1) {
        for (unsigned i = tx; i < NYD; i += TPB) {
          const unsigned ixj = i ^ j;
          if (ixj > i) {
            const float a = u_sh[i], b2 = u_sh[ixj];
            const bool up = ((i & k) == 0);
            if (up ? (a < b2) : (a > b2)) { u_sh[i] = b2; u_sh[ixj] = a; }
          }
        }
        __syncthreads();
      }
    }
    // inclusive cumsum of u_sh -> css_sh (4 elems/thread, two-level scan)
    {
      const float x0 = u_sh[4 * tx], x1 = u_sh[4 * tx + 1];
      const float x2 = u_sh[4 * tx + 2], x3 = u_sh[4 * tx + 3];
      const float l1 = x0 + x1, l2 = l1 + x2, l3 = l2 + x3;
      float s = l3;
#pragma unroll
      for (int off = 1; off < 32; off <<= 1) {
        const float y = __shfl_up(s, off, 32);
        if (lane >= off) s += y;
      }
      if (lane == 31) wt_sh[wv] = s;
      __syncthreads();
      if (tx == 0) {
        float a = 0.f;
        for (int i = 0; i < 8; ++i) { const float tmp = wt_sh[i]; wt_sh[i] = a; a += tmp; }
      }
      __syncthreads();
      const float excl = wt_sh[wv] + (s - l3);
      css_sh[4 * tx]     = excl + x0;
      css_sh[4 * tx + 1] = excl + l1;
      css_sh[4 * tx + 2] = excl + l2;
      css_sh[4 * tx + 3] = excl + l3;
    }
    __syncthreads();
    {
      int cnt4 = 0;
#pragma unroll
      for (int m = 0; m < 4; ++m) {
        const int i = 4 * tx + m;
        if (u_sh[i] - (css_sh[i] - 1.f) / (float)(i + 1) > 0.f) cnt4++;
      }
      atomicAdd(&rho_sh, cnt4);
    }
    __syncthreads();
    {
      const int rho = rho_sh;
      const float theta = (css_sh[rho - 1] - 1.f) / (float)rho;
      __syncthreads();
      for (int j = tx; j < NYD; j += TPB)
        z_sh[j] = fmaxf(z_sh[j] - theta, 0.f);
    }
    __syncthreads();
  }

  // write z_star
  if (blk == 0)
    for (int j = tx; j < NYD; j += TPB) z_out[j] = z_sh[j];
}

// ---------------------------------------------------------------------------
extern "C" void kernel_launch(void* const* d_in, const int* in_sizes, int n_in,
                              void* d_out, int out_size, void* d_ws, size_t ws_size,
                              hipStream_t stream) {
  (void)in_sizes; (void)n_in; (void)out_size;
  const float* X     = (const float*)d_in[0];   // (4097, 512)
  const float* Y     = (const float*)d_in[1];   // (4096, 1024)
  const float* W     = (const float*)d_in[2];   // (1024, 512)
  const float* bias  = (const float*)d_in[3];   // (1024,)
  const float* delta = (const float*)d_in[4];   // (1,)
  const float* gamma = (const float*)d_in[5];   // (1,)
  float* out = (float*)d_out;                   // [z_star(1024) | y_hat(1024)]

  float* ws   = (float*)d_ws;
  float* ep   = ws;                                   // 4096*1024
  float* epT  = ws + (size_t)NOBS * NYD;              // 1024*4096
  float* r_g  = epT + (size_t)NOBS * NYD;             // 4096
  float* v_g  = r_g + NOBS;                           // 1024
  float* yh   = v_g + NYD;                            // 1024
  float* part = yh + NYD;                             // NB*8
  unsigned* bar = (unsigned*)(part + NB * 8);         // 2 ints
  const size_t need = ((size_t)(part + NB * 8) - (size_t)ws) + 2 * sizeof(unsigned);
  if (ws_size < need) return;  // cannot proceed without scratch

  hipMemsetAsync(bar, 0, 2 * sizeof(unsigned), stream);
  gemm_resid_kernel<<<514, 256, 0, stream>>>(X, Y, W, bias, ep, yh, out + NYD);
  transpose_kernel<<<dim3(NYD / 32, NOBS / 32), 256, 0, stream>>>(ep, epT);
  solve_kernel<<<NB, TPB, 0, stream>>>(ep, epT, r_g, v_g, yh, part, bar,
                                       delta, gamma, out);
}